// SAGELayer_55783035240592
// MI455X (gfx1250) — compile-verified
//
#include <hip/hip_runtime.h>

typedef float v2f __attribute__((ext_vector_type(2)));
typedef float v8f __attribute__((ext_vector_type(8)));

#define F 64
#define TILE_M 16
#define WAVES_PER_BLOCK 4
#define WPITCH 65   // padded pitch for transposed weights in LDS
#define MPITCH 66   // padded (even) pitch for mean tile in LDS

// ---------------- small utility kernels ----------------

__global__ void zero_i32_kernel(int* __restrict__ p, int n) {
    int i = blockIdx.x * blockDim.x + threadIdx.x;
    if (i < n) p[i] = 0;
}

__global__ void count_kernel(const int* __restrict__ dst, int* __restrict__ cnt, int E) {
    int e = blockIdx.x * blockDim.x + threadIdx.x;
    if (e < E) atomicAdd(&cnt[dst[e]], 1);
}

// Single-workgroup exclusive prefix sum over n counts -> offsets[0..n]
__global__ void scan_kernel(const int* __restrict__ cnt, int* __restrict__ offsets, int n) {
    __shared__ int tmp[1024];
    __shared__ int carry;
    int t = threadIdx.x;
    if (t == 0) carry = 0;
    __syncthreads();
    for (int base = 0; base < n; base += 1024) {
        int i = base + t;
        int v = (i < n) ? cnt[i] : 0;
        tmp[t] = v;
        __syncthreads();
        for (int off = 1; off < 1024; off <<= 1) {
            int add = (t >= off) ? tmp[t - off] : 0;
            __syncthreads();
            tmp[t] += add;
            __syncthreads();
        }
        int incl = tmp[t];
        int c = carry;
        if (i < n) offsets[i] = c + incl - v;   // exclusive
        __syncthreads();
        if (t == 1023) carry = c + incl;        // running total
        __syncthreads();
    }
    if (t == 0) offsets[n] = carry;
}

__global__ void scatter_kernel(const int* __restrict__ src, const int* __restrict__ dst,
                               const int* __restrict__ offsets, int* __restrict__ cur,
                               int* __restrict__ csr_src, int E) {
    int e = blockIdx.x * blockDim.x + threadIdx.x;
    if (e < E) {
        int d = dst[e];
        int p = atomicAdd(&cur[d], 1);
        csr_src[offsets[d] + p] = src[e];
    }
}

// ---------------- fused mean-aggregate + dual GEMM (WMMA f32) + bias + ReLU ----------------
// One wave per 16-node tile. LDS: transposed W_l/W_r + per-wave mean tile.

__global__ __launch_bounds__(32 * WAVES_PER_BLOCK) void sage_fused_kernel(
    const float* __restrict__ x, const int* __restrict__ offsets,
    const int* __restrict__ csr_src, const float* __restrict__ Wl,
    const float* __restrict__ bl, const float* __restrict__ Wr,
    float* __restrict__ out, int nTiles, int N) {

    __shared__ float sWl[F * WPITCH];                          // WlT[k][n] = Wl[n][k]
    __shared__ float sWr[F * WPITCH];
    __shared__ float sMean[WAVES_PER_BLOCK * TILE_M * MPITCH]; // per-wave 16x64 mean tile

    // cooperative transpose-load of both weight matrices (coalesced global reads)
    for (int i = threadIdx.x; i < F * F; i += blockDim.x) {
        int n = i >> 6, k = i & 63;
        sWl[k * WPITCH + n] = Wl[i];
        sWr[k * WPITCH + n] = Wr[i];
    }
    __syncthreads();

    int lane = threadIdx.x & 31;
    int wv   = threadIdx.x >> 5;
    int tile = blockIdx.x * WAVES_PER_BLOCK + wv;
    bool valid = tile < nTiles;
    int row0 = tile * TILE_M;
    float* myMean = &sMean[wv * TILE_M * MPITCH];

    // ---- Phase A: mean aggregation. Each lane owns 2 features (float2). ----
    if (valid) {
        for (int m = 0; m < TILE_M; ++m) {
            int node = row0 + m;
            float ax = 0.f, ay = 0.f;
            int deg = 0;
            if (node < N) {
                int beg = offsets[node];
                int end = offsets[node + 1];
                deg = end - beg;
                for (int e = beg; e < end; ++e) {
                    int s = csr_src[e];                       // wave-uniform
                    v2f v = *(const v2f*)(x + (size_t)s * F + 2 * lane); // coalesced 256B
                    ax += v.x; ay += v.y;
                }
            }
            float inv = 1.0f / fmaxf((float)deg, 1.0f);
            v2f mv; mv.x = ax * inv; mv.y = ay * inv;
            *(v2f*)(myMean + m * MPITCH + 2 * lane) = mv;
        }
    }
    __syncthreads();
    if (!valid) return;

    // ---- Phase B: out = mean*WlT + x*WrT + b, ReLU, via V_WMMA_F32_16X16X4_F32 ----
    // A-frag (16x4 f32): lanes 0-15 -> M=lane, K={0,1}; lanes 16-31 -> M=lane-16, K={2,3}
    int arow = lane & 15;
    int krow = (lane >> 4) * 2;
    int xr = row0 + arow; if (xr >= N) xr = N - 1;            // clamp (partial tile safety)
    const float* xrow = x + (size_t)xr * F;

    for (int ct = 0; ct < 4; ++ct) {
        int nn = ct * 16 + arow;                               // output column for B/C frags
        v8f acc = {};
        for (int kb = 0; kb < 16; ++kb) {
            int k = kb * 4 + krow;
            v2f a = *(const v2f*)(myMean + arow * MPITCH + k);
            v2f b;
            b.x = sWl[k * WPITCH + nn];
            b.y = sWl[(k + 1) * WPITCH + nn];
            acc = __builtin_amdgcn_wmma_f32_16x16x4_f32(false, a, false, b,
                                                        (short)0, acc, false, false);
        }
        for (int kb = 0; kb < 16; ++kb) {
            int k = kb * 4 + krow;
            v2f a = *(const v2f*)(xrow + k);
            v2f b;
            b.x = sWr[k * WPITCH + nn];
            b.y = sWr[(k + 1) * WPITCH + nn];
            acc = __builtin_amdgcn_wmma_f32_16x16x4_f32(false, a, false, b,
                                                        (short)0, acc, false, false);
        }
        float bias = bl[nn];
        int mbase = (lane >> 4) * 8;                           // C/D frag: VGPR r -> M = mbase+r
        for (int r = 0; r < 8; ++r) {
            int row = row0 + mbase + r;
            if (row < N) {
                float v = acc[r] + bias;
                out[(size_t)row * F + nn] = fmaxf(v, 0.0f);
            }
        }
    }
}

// ---------------- launcher ----------------

extern "C" void kernel_launch(void* const* d_in, const int* in_sizes, int n_in,
                              void* d_out, int out_size, void* d_ws, size_t ws_size,
                              hipStream_t stream) {
    const float* x   = (const float*)d_in[0];
    const int*   ei  = (const int*)d_in[1];
    const float* Wl  = (const float*)d_in[2];
    const float* bl  = (const float*)d_in[3];
    const float* Wr  = (const float*)d_in[4];
    float*       out = (float*)d_out;

    int N = in_sizes[0] / F;
    int E = in_sizes[1] / 2;
    const int* src = ei;          // edge_index[0]
    const int* dst = ei + E;      // edge_index[1]

    // workspace layout: cnt[N] | cur[N] | offsets[N+1] | csr_src[E]
    int* cnt     = (int*)d_ws;
    int* cur     = cnt + N;
    int* offsets = cur + N;
    int* csr_src = offsets + (N + 1);

    zero_i32_kernel<<<(2 * N + 255) / 256, 256, 0, stream>>>(cnt, 2 * N);
    count_kernel<<<(E + 255) / 256, 256, 0, stream>>>(dst, cnt, E);
    scan_kernel<<<1, 1024, 0, stream>>>(cnt, offsets, N);
    scatter_kernel<<<(E + 255) / 256, 256, 0, stream>>>(src, dst, offsets, cur, csr_src, E);

    int nTiles = (N + TILE_M - 1) / TILE_M;
    int nBlocks = (nTiles + WAVES_PER_BLOCK - 1) / WAVES_PER_BLOCK;
    sage_fused_kernel<<<nBlocks, 32 * WAVES_PER_BLOCK, 0, stream>>>(
        x, offsets, csr_src, Wl, bl, Wr, out, nTiles, N);
}